// MACENet_81200651698608
// MI455X (gfx1250) — compile-verified
//
#include <hip/hip_runtime.h>
#include <hip/hip_bf16.h>
#include <math.h>

// ---------------------------------------------------------------------------
// MACE 2-layer forward for MI455X (gfx1250, wave32, WMMA).
//
// Sizes: N=8192 nodes, E=131072 edges, C=64 channels, 16 SH components,
// S=10 species, NB=8 bessel, HID=64, HREAD=16.
// ---------------------------------------------------------------------------

#define N_NODES 8192
#define N_EDGES 131072
#define CCH     64
#define NSPEC   10

typedef float v2f __attribute__((ext_vector_type(2)));
typedef float v8f __attribute__((ext_vector_type(8)));

// ---------------------------------------------------------------------------
// zero workspace
// ---------------------------------------------------------------------------
__global__ void zero_f32(float* __restrict__ p, int n) {
    int i = blockIdx.x * 256 + threadIdx.x;
    if (i < n) p[i] = 0.0f;
}

// ---------------------------------------------------------------------------
// s0[n,c] = W_embed[node_z[n], c]
// ---------------------------------------------------------------------------
__global__ void embed_s0(const int* __restrict__ node_z,
                         const float* __restrict__ W_embed,
                         float* __restrict__ s0) {
    int i = blockIdx.x * 256 + threadIdx.x;   // over N*64
    int n = i >> 6, c = i & 63;
    s0[i] = W_embed[node_z[n] * CCH + c];
}

// ---------------------------------------------------------------------------
// Fused edge kernel: spherical harmonics + bessel*envelope + radial MLP
// (silu(remb@Wr1) @ Wr2) + message formation + atomic scatter into agg.
// Block = 256 threads = 4 edges x 64 channels. h (HID=64) shared via LDS.
// ---------------------------------------------------------------------------
__global__ void edge_scatter(const float* __restrict__ vectors,
                             const int* __restrict__ senders,
                             const int* __restrict__ receivers,
                             const float* __restrict__ sfeat,   // [N,64]
                             const float* __restrict__ Wr1,     // [8,64]
                             const float* __restrict__ Wr2,     // [64,256]
                             float* __restrict__ agg) {         // [N,16,64]
    __shared__ float hbuf[4][CCH];
    const int tid = threadIdx.x;
    const int g   = tid >> 6;           // edge slot in block
    const int c   = tid & 63;           // channel
    const int e   = blockIdx.x * 4 + g; // E divisible by 4

    // --- geometry (redundant per lane; cheap, stays in registers) ---
    float vx = vectors[e*3+0], vy = vectors[e*3+1], vz = vectors[e*3+2];
    float r  = sqrtf(vx*vx + vy*vy + vz*vz + 1e-12f);
    float ir = 1.0f / r;
    float x = vx*ir, y = vy*ir, z = vz*ir;

    // radial embedding: bessel * polynomial envelope (RMAX=5, p=6)
    float xr = r * 0.2f;
    float env = 0.0f;
    if (xr < 1.0f) {
        float x3 = xr*xr*xr, x6 = x3*x3, x7 = x6*xr, x8 = x7*xr;
        env = 1.0f - 28.0f*x6 + 48.0f*x7 - 21.0f*x8;
    }
    float pref = 0.6324555320336759f * ir * env;   // sqrt(2/5)/r * env
    float remb[8];
    #pragma unroll
    for (int k = 0; k < 8; ++k)
        remb[k] = pref * __sinf((float)(k+1) * 3.14159265358979323846f * xr);

    // h[c] = silu(remb @ Wr1) -> LDS
    float acc = 0.0f;
    #pragma unroll
    for (int k = 0; k < 8; ++k) acc += remb[k] * Wr1[k*CCH + c];
    hbuf[g][c] = acc / (1.0f + __expf(-acc));
    __syncthreads();

    // rw[l][c] = sum_j h[j] * Wr2[j, l*64+c]   (Wr2 L2-resident, 64KB)
    float rw0 = 0.f, rw1 = 0.f, rw2 = 0.f, rw3 = 0.f;
    for (int j = 0; j < CCH; ++j) {
        float hj = hbuf[g][j];
        const float* w = Wr2 + j*256 + c;
        rw0 += hj * w[0];
        rw1 += hj * w[64];
        rw2 += hj * w[128];
        rw3 += hj * w[192];
    }

    // real spherical harmonics up to l=3 (component normalization)
    const float s3  = 1.7320508075688772f;
    const float s5  = 2.2360679774997896f;
    const float s15 = 3.8729833462074170f;
    const float s7  = 2.6457513110645906f;
    const float s105= 10.246950765959598f;
    const float c358= 2.0916500663351889f;   // sqrt(35/8)
    const float c218= 1.6201851746019651f;   // sqrt(21/8)
    float zz = z*z, xx = x*x, yy = y*y;
    float Y[16];
    Y[0]  = 1.0f;
    Y[1]  = s3*x;  Y[2] = s3*y;  Y[3] = s3*z;
    Y[4]  = s15*x*y;
    Y[5]  = s15*y*z;
    Y[6]  = 0.5f*s5*(3.0f*zz - 1.0f);
    Y[7]  = s15*x*z;
    Y[8]  = 0.5f*s15*(xx - yy);
    Y[9]  = c358*y*(3.0f*xx - yy);
    Y[10] = s105*x*y*z;
    Y[11] = c218*y*(5.0f*zz - 1.0f);
    Y[12] = 0.5f*s7*z*(5.0f*zz - 3.0f);
    Y[13] = c218*x*(5.0f*zz - 1.0f);
    Y[14] = 0.5f*s105*z*(xx - yy);
    Y[15] = c358*x*(xx - 3.0f*yy);

    const float rwl[4] = {rw0, rw1, rw2, rw3};
    const int LM[16] = {0,1,1,1,2,2,2,2,2,3,3,3,3,3,3,3};

    int   snd = senders[e];
    int   rcv = receivers[e];
    float sc  = sfeat[snd*CCH + c];
    float* base = agg + (size_t)rcv * 1024 + c;
    #pragma unroll
    for (int m = 0; m < 16; ++m)
        atomicAdd(base + m*CCH, sc * Y[m] * rwl[LM[m]]);
}

// ---------------------------------------------------------------------------
// Node stage after interaction 1. Only the m=0 slice survives downstream:
//   t   = (agg[n,0,:]/16) @ Wmix0[0]
//   f0  = t @ Wsel[z,0]
//   s1  = (wp0 + wp1*f0 + wp2*f0^2) * f0
//   e0  = s1 . Wread0 ;  resid = s1 @ Wres[z]
// Block = 256 = 4 nodes x 64 channels.
// ---------------------------------------------------------------------------
__global__ void node_stage1(const float* __restrict__ agg,
                            const int* __restrict__ node_z,
                            const float* __restrict__ Wmix0,   // [4,64,64]
                            const float* __restrict__ Wsel,    // [S,4,64,64]
                            const float* __restrict__ wprod0,  // [3,S,64]
                            const float* __restrict__ Wres,    // [S,64,64]
                            const float* __restrict__ Wread0,  // [64]
                            float* __restrict__ s1,
                            float* __restrict__ resid,
                            float* __restrict__ e0) {
    __shared__ float sA[4][CCH], sT[4][CCH], sS1[4][CCH], sRed[4][CCH];
    const int tid = threadIdx.x;
    const int g = tid >> 6, c = tid & 63;
    const int n = blockIdx.x * 4 + g;

    sA[g][c] = agg[(size_t)n * 1024 + c] * 0.0625f;   // / AVG_NEIGH
    __syncthreads();

    float t = 0.f;
    for (int j = 0; j < CCH; ++j) t += sA[g][j] * Wmix0[j*CCH + c];  // Wmix0[0]
    sT[g][c] = t;
    __syncthreads();

    const int z = node_z[n];
    const float* Ws = Wsel + (size_t)z * (4*CCH*CCH);                // Wsel[z,0]
    float f0 = 0.f;
    for (int j = 0; j < CCH; ++j) f0 += sT[g][j] * Ws[j*CCH + c];

    float w0 = wprod0[(0*NSPEC + z)*CCH + c];
    float w1 = wprod0[(1*NSPEC + z)*CCH + c];
    float w2 = wprod0[(2*NSPEC + z)*CCH + c];
    float s1c = (w0 + w1*f0 + w2*f0*f0) * f0;

    s1[n*CCH + c]  = s1c;
    sS1[g][c]      = s1c;
    sRed[g][c]     = s1c * Wread0[c];
    __syncthreads();

    if (c == 0) {
        float e = 0.f;
        for (int j = 0; j < CCH; ++j) e += sRed[g][j];
        e0[n] = e;
    }
    const float* Wr = Wres + (size_t)z * (CCH*CCH);
    float rs = 0.f;
    for (int j = 0; j < CCH; ++j) rs += sS1[g][j] * Wr[j*CCH + c];
    resid[n*CCH + c] = rs;
}

// ---------------------------------------------------------------------------
// Final node stage: full 16-component mix via V_WMMA_F32_16X16X4_F32,
// then invariants + power series + silu readout -> node energies.
// Block = 256 threads (8 waves) per 16-node tile. Each wave owns 2 m-values.
// f_lds[16][16][64] = 64KB; f_lds[0]/f_lds[1] reused for B / h in epilogue.
//
// WMMA fragment layouts (ISA 7.12.2):
//   A 16x4 f32 : lane<16 -> M=lane, K={0,1}; lane>=16 -> M=lane-16, K={2,3}
//   B 4x16 f32 : lane<16 -> N=lane, K={0,1}; lane>=16 -> N=lane-16, K={2,3}
//   D 16x16 f32: VGPR v -> M = v + 8*(lane>=16), N = lane&15
// ---------------------------------------------------------------------------
__global__ void node_stage2_wmma(const float* __restrict__ agg,
                                 const int* __restrict__ node_z,
                                 const float* __restrict__ Wmix1,   // [4,64,64]
                                 const float* __restrict__ wpow,    // [3,S,64]
                                 const float* __restrict__ winv,    // [4,S,64]
                                 const float* __restrict__ resid,   // [N,64]
                                 const float* __restrict__ e0,      // [N]
                                 const float* __restrict__ Wread1a, // [64,16]
                                 const float* __restrict__ Wread1b, // [16]
                                 const float* __restrict__ atomic_E,// [S]
                                 float* __restrict__ out) {         // [N]
    __shared__ float f_lds[16][16][CCH];   // [m][node][c], 64KB
    const int tid  = threadIdx.x;
    const int wave = tid >> 5;
    const int lane = tid & 31;
    const int hl   = lane >> 4;            // 0 / 1 half-wave
    const int l16  = lane & 15;
    const int nb   = blockIdx.x * 16;      // node tile base

    for (int mi = 0; mi < 2; ++mi) {
        const int m  = wave * 2 + mi;
        const int lm = (m == 0) ? 0 : ((m < 4) ? 1 : ((m < 9) ? 2 : 3));
        const float* B = Wmix1 + lm * (CCH*CCH);

        // preload A fragments for this m (reused across 4 N-tiles)
        v2f a[16];
        const float* Arow = agg + (size_t)(nb + l16) * 1024 + m * CCH;
        #pragma unroll
        for (int kk = 0; kk < 16; ++kk) {
            int k0 = kk*4 + hl*2;
            a[kk].x = Arow[k0]     * 0.0625f;
            a[kk].y = Arow[k0 + 1] * 0.0625f;
        }
        for (int nt = 0; nt < 4; ++nt) {
            v8f acc = {};
            const int col = nt*16 + l16;
            #pragma unroll
            for (int kk = 0; kk < 16; ++kk) {
                int k0 = kk*4 + hl*2;
                v2f b;
                b.x = B[k0*CCH + col];
                b.y = B[(k0+1)*CCH + col];
                acc = __builtin_amdgcn_wmma_f32_16x16x4_f32(
                          false, a[kk], false, b, (short)0, acc, false, false);
            }
            #pragma unroll
            for (int v = 0; v < 8; ++v)
                f_lds[m][v + hl*8][col] = acc[v];
        }
    }
    __syncthreads();

    // --- phase A: invariants + power series + resid -> B (reuse f_lds[0]) ---
    float* B_lds = &f_lds[0][0][0];
    for (int i = 0; i < 4; ++i) {
        const int p = tid + i*256;           // 1024 (node,c) pairs
        const int node = p >> 6, c = p & 63;
        float sc = f_lds[0][node][c];
        float inv0 = sc*sc, inv1 = 0.f, inv2 = 0.f, inv3 = 0.f;
        #pragma unroll
        for (int m = 1; m < 4; ++m)  { float v = f_lds[m][node][c]; inv1 += v*v; }
        #pragma unroll
        for (int m = 4; m < 9; ++m)  { float v = f_lds[m][node][c]; inv2 += v*v; }
        #pragma unroll
        for (int m = 9; m < 16; ++m) { float v = f_lds[m][node][c]; inv3 += v*v; }
        const int z = node_z[nb + node];
        float p1 = wpow[(0*NSPEC + z)*CCH + c];
        float p2 = wpow[(1*NSPEC + z)*CCH + c];
        float p3 = wpow[(2*NSPEC + z)*CCH + c];
        float i0 = winv[(0*NSPEC + z)*CCH + c];
        float i1 = winv[(1*NSPEC + z)*CCH + c];
        float i2 = winv[(2*NSPEC + z)*CCH + c];
        float i3 = winv[(3*NSPEC + z)*CCH + c];
        float Bv = p1*sc + p2*sc*sc + p3*sc*sc*sc
                 + i0*inv0 + i1*inv1 + i2*inv2 + i3*inv3
                 + resid[(size_t)(nb + node)*CCH + c];
        B_lds[node*CCH + c] = Bv;            // owner-only read/write: race-free
    }
    __syncthreads();

    // --- phase B: h = silu(B @ Wread1a) * Wread1b  (16 nodes x 16 hidden) ---
    {
        const int node = tid >> 4, j = tid & 15;
        float hv = 0.f;
        for (int cc = 0; cc < CCH; ++cc)
            hv += B_lds[node*CCH + cc] * Wread1a[cc*16 + j];
        hv = hv / (1.0f + __expf(-hv));
        float* h_lds = &f_lds[1][0][0];      // f_lds[1] region now free
        h_lds[node*16 + j] = hv * Wread1b[j];
    }
    __syncthreads();

    // --- phase C: e1 reduce + final energy ---
    if (tid < 16) {
        const int node = tid;
        const float* h_lds = &f_lds[1][0][0];
        float e1 = 0.f;
        for (int j = 0; j < 16; ++j) e1 += h_lds[node*16 + j];
        const int z = node_z[nb + node];
        // MEAN=0, STD=1
        out[nb + node] = (e0[nb + node] + e1) + atomic_E[z];
    }
}

// ---------------------------------------------------------------------------
// launch
// ---------------------------------------------------------------------------
extern "C" void kernel_launch(void* const* d_in, const int* in_sizes, int n_in,
                              void* d_out, int out_size, void* d_ws, size_t ws_size,
                              hipStream_t stream) {
    const float* vectors   = (const float*)d_in[0];
    const int*   node_z    = (const int*)  d_in[1];
    const int*   senders   = (const int*)  d_in[2];
    const int*   receivers = (const int*)  d_in[3];
    const float* W_embed   = (const float*)d_in[4];
    const float* Wr1_0     = (const float*)d_in[5];
    const float* Wr2_0     = (const float*)d_in[6];
    const float* Wmix0     = (const float*)d_in[7];
    const float* Wsel      = (const float*)d_in[8];
    const float* wprod0    = (const float*)d_in[9];
    const float* Wr1_1     = (const float*)d_in[10];
    const float* Wr2_1     = (const float*)d_in[11];
    const float* Wmix1     = (const float*)d_in[12];
    const float* Wres      = (const float*)d_in[13];
    const float* wpow      = (const float*)d_in[14];
    const float* winv      = (const float*)d_in[15];
    const float* Wread0    = (const float*)d_in[16];
    const float* Wread1a   = (const float*)d_in[17];
    const float* Wread1b   = (const float*)d_in[18];
    const float* atomic_E  = (const float*)d_in[19];
    float* out = (float*)d_out;

    // workspace layout (floats)
    float* ws    = (float*)d_ws;
    float* agg   = ws;                         // N*16*64   = 8388608
    float* s0    = agg   + (size_t)N_NODES*1024;   // 524288
    float* s1    = s0    + (size_t)N_NODES*CCH;    // 524288
    float* resid = s1    + (size_t)N_NODES*CCH;    // 524288
    float* e0    = resid + (size_t)N_NODES*CCH;    // 8192

    const int AGG_ELEMS = N_NODES * 1024;

    // ---- interaction 1 ----
    zero_f32<<<AGG_ELEMS/256, 256, 0, stream>>>(agg, AGG_ELEMS);
    embed_s0<<<(N_NODES*CCH)/256, 256, 0, stream>>>(node_z, W_embed, s0);
    edge_scatter<<<N_EDGES/4, 256, 0, stream>>>(vectors, senders, receivers,
                                                s0, Wr1_0, Wr2_0, agg);
    node_stage1<<<N_NODES/4, 256, 0, stream>>>(agg, node_z, Wmix0, Wsel, wprod0,
                                               Wres, Wread0, s1, resid, e0);
    // ---- interaction 2 ----
    zero_f32<<<AGG_ELEMS/256, 256, 0, stream>>>(agg, AGG_ELEMS);
    edge_scatter<<<N_EDGES/4, 256, 0, stream>>>(vectors, senders, receivers,
                                                s1, Wr1_1, Wr2_1, agg);
    node_stage2_wmma<<<N_NODES/16, 256, 0, stream>>>(agg, node_z, Wmix1, wpow, winv,
                                                     resid, e0, Wread1a, Wread1b,
                                                     atomic_E, out);
}